// HashGrid4D_39427799777696
// MI455X (gfx1250) — compile-verified
//
#include <hip/hip_runtime.h>
#include <cstdint>

// Multi-res 4D hash-grid encode (Instant-NGP / hexplane style) for MI455X.
// Layout: one thread per (point b, unit) where unit = (table<<4)|level,
// 64 units per point. Wave32-friendly: lanes of a wave cover 32 consecutive
// units of (at most) two tables; all control flow is select-based (no
// divergence). Each thread performs 8 independent global_load_b64 gathers
// issued before any use -> deep MLP per wave to hide HBM/L2 latency.

typedef __attribute__((ext_vector_type(2))) float v2f;

#define HG_P1 2654435761u
#define HG_P2 805459861u
#define HG_HASH_MASK 0x7FFFFu   // all hashed levels have hsize = 2^19

__global__ __launch_bounds__(256) void hashgrid4d_kernel(
    const float* __restrict__ xin,   // [B,3]
    const float* __restrict__ tin,   // [B]
    const float* __restrict__ e0,    // emb_static  [TABLE_SIZE,2]
    const float* __restrict__ e1,    // emb_xyt
    const float* __restrict__ e2,    // emb_xzt
    const float* __restrict__ e3,    // emb_yzt
    float* __restrict__ out,         // [B,128]
    int B)
{
    const int gid  = blockIdx.x * 256 + threadIdx.x;
    const int b    = gid >> 6;            // point index
    if (b >= B) return;
    const int unit = gid & 63;            // (table<<4) | level
    const int tbl  = unit >> 4;
    const int lvl  = unit & 15;

    // ---- per-table plane coordinates ------------------------------------
    // tbl 0: (x,y,z)  tbl 1: (x,y,t)  tbl 2: (x,z,t)  tbl 3: (y,z,t)
    const float x0 = xin[3 * b + 0];
    const float x1 = xin[3 * b + 1];
    const float x2 = xin[3 * b + 2];
    const float tt = tin[b];
    const float p0 = (tbl == 3) ? x1 : x0;
    const float p1 = (tbl >= 2) ? x2 : x1;
    const float p2 = (tbl == 0) ? x2 : tt;

    // ---- grid coords + trilinear fractions ------------------------------
    const uint32_t res = 16u << lvl;
    const float rf  = (float)res;
    const float f0 = p0 * rf, f1 = p1 * rf, f2 = p2 * rf;
    const float fl0 = floorf(f0), fl1 = floorf(f1), fl2 = floorf(f2);
    const float fr0 = f0 - fl0, fr1 = f1 - fl1, fr2 = f2 - fl2;  // pre-clip frac (matches ref)
    const float rm1 = rf - 1.0f;
    const uint32_t g0 = (uint32_t)fminf(fmaxf(fl0, 0.0f), rm1);
    const uint32_t g1 = (uint32_t)fminf(fmaxf(fl1, 0.0f), rm1);
    const uint32_t g2 = (uint32_t)fminf(fmaxf(fl2, 0.0f), rm1);

    // ---- level base offset + indexing mode ------------------------------
    // dense levels: 0 (17^3=4913), 1 (33^3=35937), 2 (65^3=274625)
    // hashed levels 3..15: 2^19 entries each; offsets = 315475 + (l-3)*2^19
    const bool dense = (lvl <= 2);
    const uint32_t r1 = res + 1u;
    const uint32_t off = (lvl >= 3) ? (315475u + ((uint32_t)(lvl - 3) << 19))
                        : (lvl == 2) ? 40850u
                        : (lvl == 1) ? 4913u : 0u;
    const float* base =
        ((tbl == 0) ? e0 : (tbl == 1) ? e1 : (tbl == 2) ? e2 : e3)
        + (size_t)off * 2u;

    // ---- compute all 8 corner indices first (branchless select) ---------
    uint32_t idx[8];
#pragma unroll
    for (int c = 0; c < 8; ++c) {
        const uint32_t h0 = g0 + (uint32_t)(c & 1);
        const uint32_t h1 = g1 + (uint32_t)((c >> 1) & 1);
        const uint32_t h2 = g2 + (uint32_t)(c >> 2);
        const uint32_t id_dense = h0 + h1 * r1 + h2 * r1 * r1;   // valid only when dense
        const uint32_t id_hash  = (h0 ^ (h1 * HG_P1) ^ (h2 * HG_P2)) & HG_HASH_MASK;
        idx[c] = dense ? id_dense : id_hash;
    }

    // ---- issue all 8 gathers back-to-back (8-deep loadcnt per lane) -----
    v2f val[8];
#pragma unroll
    for (int c = 0; c < 8; ++c)
        val[c] = *(const v2f*)(base + (size_t)idx[c] * 2u);

    // ---- trilinear accumulate ------------------------------------------
    const float wa0 = 1.0f - fr0, wa1 = 1.0f - fr1, wa2 = 1.0f - fr2;
    v2f acc = (v2f)(0.0f);
#pragma unroll
    for (int c = 0; c < 8; ++c) {
        const float w = ((c & 1) ? fr0 : wa0)
                      * ((c & 2) ? fr1 : wa1)
                      * ((c & 4) ? fr2 : wa2);
        acc += w * val[c];
    }

    // ---- streamed NT store: keep the 192MB L2 for the hash tables -------
    v2f* dst = (v2f*)(out + (size_t)b * 128u + (size_t)unit * 2u);
    __builtin_nontemporal_store(acc, dst);
}

extern "C" void kernel_launch(void* const* d_in, const int* in_sizes, int n_in,
                              void* d_out, int out_size, void* d_ws, size_t ws_size,
                              hipStream_t stream) {
    const float* x  = (const float*)d_in[0];
    const float* t  = (const float*)d_in[1];
    const float* e0 = (const float*)d_in[2];
    const float* e1 = (const float*)d_in[3];
    const float* e2 = (const float*)d_in[4];
    const float* e3 = (const float*)d_in[5];
    float* out = (float*)d_out;

    const int B = in_sizes[0] / 3;              // x is [B,3]
    const long long total = (long long)B * 64;  // one thread per (point, table, level)
    const int block = 256;
    const long long grid = (total + block - 1) / block;

    hashgrid4d_kernel<<<(unsigned)grid, block, 0, stream>>>(x, t, e0, e1, e2, e3, out, B);
}